// MaskProcessor_1202590843781
// MI455X (gfx1250) — compile-verified
//
#include <hip/hip_runtime.h>

// CDNA5 / gfx1250 wave32 WMMA types
typedef __attribute__((ext_vector_type(2))) float v2f;
typedef __attribute__((ext_vector_type(8))) float v8f;

#define MAXH 512
#define MAXW 512
#define MHW  28     // mask H/W
#define KP   32     // K padded to 32 (rows/cols 28..31 are zero)

// out = Wy (512x28) @ M (28x28) @ Wx^T (28x512), per instance.
// Wy/Wx are the sparse bilinear-tap matrices (<=2 nonzeros/row); zero padding
// of M to 32x32 makes grid_sample's padding_mode='zeros' semantics exact.
//
// Block: 256 threads (8 waves), handles one 16-row x 512-col output slab.
// Grid: (N_INST, 32).
__global__ __launch_bounds__(256)
void paste_masks_wmma(const float* __restrict__ masks,
                      const float* __restrict__ boxes,
                      float* __restrict__ out)
{
    __shared__ float sM[KP * KP];   // zero-padded mask, row-major 32x32
    __shared__ float sT[16 * KP];   // intermediate T = Wy_tile @ M, 16x32

    const int n     = blockIdx.x;        // instance
    const int ytile = blockIdx.y;        // 0..31 (16 output rows each)
    const int tid   = threadIdx.x;
    const int lane  = tid & 31;
    const int wave  = tid >> 5;

    // Box -> sampling transform (matches reference exactly):
    // coord = ((p + 0.5 - b0) / (b1 - b0)) * 28 - 0.5
    const float bx0 = boxes[n * 4 + 0];
    const float by0 = boxes[n * 4 + 1];
    const float bx1 = boxes[n * 4 + 2] + 1e-5f;
    const float by1 = boxes[n * 4 + 3] + 1e-5f;
    const float sxr = (float)MHW / (bx1 - bx0);
    const float syr = (float)MHW / (by1 - by0);

    // Load mask into LDS, zero-padded to 32x32.
    const float* msrc = masks + (size_t)n * MHW * MHW;
    for (int i = tid; i < KP * KP; i += 256) {
        const int r = i >> 5, c = i & 31;
        sM[i] = (r < MHW && c < MHW) ? msrc[r * MHW + c] : 0.0f;
    }
    __syncthreads();

    // Per-lane fragment geometry (wave32 f32 WMMA):
    //   A 16x4 layout: lanes 0-15 hold M=lane, K={kk,kk+1}; lanes 16-31 hold
    //                  M=lane-16, K={kk+2,kk+3}.
    //   B 4x16 layout: mirrored, N = lane&15.
    //   C/D: d[v] at M = v + (lane>=16 ? 8 : 0), N = lane&15.
    const int half  = lane >> 4;           // 0 or 1
    const int idx16 = lane & 15;
    const int kbase = half << 1;           // 0 or 2
    const int mrow  = half << 3;           // 0 or 8

    // ---------------- Stage A: sT(16x32) = Wy_tile(16x32) @ sM(32x32) -------
    // Waves 0 and 1 each produce one 16x16 N-tile of sT.
    if (wave < 2) {
        const int ntile = wave;
        const int nn = ntile * 16 + idx16;

        // Row weights for output row y = ytile*16 + idx16.
        const float yc  = ((float)(ytile * 16 + idx16) + 0.5f - by0) * syr - 0.5f;
        const float y0f = floorf(yc);
        const int   y0  = (int)y0f;
        const float wy1 = yc - y0f;
        const float wy0 = 1.0f - wy1;

        v8f acc = {};
        #pragma unroll
        for (int kk = 0; kk < KP; kk += 4) {
            const int k0 = kk + kbase;
            const int k1 = k0 + 1;
            v2f a, b;
            a.x = (k0 == y0) ? wy0 : ((k0 == y0 + 1) ? wy1 : 0.0f);
            a.y = (k1 == y0) ? wy0 : ((k1 == y0 + 1) ? wy1 : 0.0f);
            b.x = sM[k0 * KP + nn];
            b.y = sM[k1 * KP + nn];
            acc = __builtin_amdgcn_wmma_f32_16x16x4_f32(
                false, a, false, b, (short)0, acc, false, false);
        }
        #pragma unroll
        for (int v = 0; v < 8; ++v)
            sT[(mrow + v) * KP + nn] = acc[v];
    }
    __syncthreads();

    // ---------------- Stage B: out_tile(16x16) = sT(16x32) @ Wx^T chunk -----
    // Preload this lane's A fragments (tmp rows) for all 8 k-steps.
    float a0[8], a1[8];
    #pragma unroll
    for (int s = 0; s < 8; ++s) {
        const int k0 = 4 * s + kbase;
        a0[s] = sT[idx16 * KP + k0];
        a1[s] = sT[idx16 * KP + k0 + 1];
    }

    float* orow = out + ((size_t)n * MAXH + (size_t)ytile * 16) * MAXW;

    #pragma unroll
    for (int t = 0; t < 4; ++t) {
        const int xtile = wave * 4 + t;          // 0..31
        const int x     = xtile * 16 + idx16;    // this lane's N column

        // Column weights for output col x.
        const float xc  = ((float)x + 0.5f - bx0) * sxr - 0.5f;
        const float x0f = floorf(xc);
        const int   x0  = (int)x0f;
        const float wx1 = xc - x0f;
        const float wx0 = 1.0f - wx1;

        v8f acc = {};
        #pragma unroll
        for (int s = 0; s < 8; ++s) {
            const int k0 = 4 * s + kbase;
            const int k1 = k0 + 1;
            v2f a, b;
            a.x = a0[s];
            a.y = a1[s];
            b.x = (k0 == x0) ? wx0 : ((k0 == x0 + 1) ? wx1 : 0.0f);
            b.y = (k1 == x0) ? wx0 : ((k1 == x0 + 1) ? wx1 : 0.0f);
            acc = __builtin_amdgcn_wmma_f32_16x16x4_f32(
                false, a, false, b, (short)0, acc, false, false);
        }

        // Store C/D tile: lane writes rows mrow..mrow+7 at column x.
        #pragma unroll
        for (int v = 0; v < 8; ++v)
            orow[(size_t)(mrow + v) * MAXW + x] = acc[v];
    }
}

extern "C" void kernel_launch(void* const* d_in, const int* in_sizes, int n_in,
                              void* d_out, int out_size, void* d_ws, size_t ws_size,
                              hipStream_t stream) {
    const float* masks = (const float*)d_in[0];  // (N,1,28,28) f32
    const float* boxes = (const float*)d_in[1];  // (N,4) f32
    float* out = (float*)d_out;                  // (N,1,512,512) f32

    const int N = in_sizes[1] / 4;               // number of instances (100)
    dim3 grid(N, MAXH / 16);                     // (100, 32)
    paste_masks_wmma<<<grid, 256, 0, stream>>>(masks, boxes, out);
}